// BuildGcn_16080357556434
// MI455X (gfx1250) — compile-verified
//
#include <hip/hip_runtime.h>

typedef float v2f __attribute__((ext_vector_type(2)));
typedef float v8f __attribute__((ext_vector_type(8)));

#define C_DIM 128  // C_IN == C_OUT == 128

// ---------------- degree / normalization ----------------

__global__ void k_init_deg(float* __restrict__ deg, int n) {
  int i = blockIdx.x * blockDim.x + threadIdx.x;
  if (i < n) deg[i] = 1.0f;  // self-loop contribution
}

__global__ void k_count_deg(const int* __restrict__ col, float* __restrict__ deg, int e) {
  int i = blockIdx.x * blockDim.x + threadIdx.x;
  if (i < e) atomicAdd(&deg[col[i]], 1.0f);
}

__global__ void k_rsqrt(const float* __restrict__ deg, float* __restrict__ dinv, int n) {
  int i = blockIdx.x * blockDim.x + threadIdx.x;
  if (i < n) {
    float d = deg[i];
    dinv[i] = (d > 0.0f) ? rsqrtf(d) : 0.0f;
  }
}

// agg[i, c] = x[i, c] * dinv[i]^2   (self-loop term; also initializes agg)
__global__ void k_selfloop(const float* __restrict__ x, const float* __restrict__ dinv,
                           float* __restrict__ agg, int total) {
  int i = blockIdx.x * blockDim.x + threadIdx.x;
  if (i < total) {
    float di = dinv[i >> 7];  // node = i / 128
    agg[i] = x[i] * di * di;
  }
}

// ---------------- edge scatter (memory/atomic bound) ----------------
// One wave32 per edge; each lane owns 4 consecutive channels (float4):
// a wave touches one contiguous 512B source row and one contiguous 512B
// destination row -> fully coalesced; accumulation via global_atomic_add_f32.
__global__ void k_scatter(const float* __restrict__ x, const int* __restrict__ row,
                          const int* __restrict__ col, const float* __restrict__ dinv,
                          float* __restrict__ agg, int e) {
  long tid = (long)blockIdx.x * blockDim.x + threadIdx.x;
  long edge = tid >> 5;
  int lane = (int)(tid & 31);
  if (edge >= e) return;
  int r = row[edge];
  int c = col[edge];
  float norm = dinv[r] * dinv[c];
  const float4* src = (const float4*)(x + (size_t)c * C_DIM);
  float4 v = src[lane];
  float* dst = agg + (size_t)r * C_DIM + lane * 4;
  atomicAdd(dst + 0, norm * v.x);
  atomicAdd(dst + 1, norm * v.y);
  atomicAdd(dst + 2, norm * v.z);
  atomicAdd(dst + 3, norm * v.w);
}

// ---------------- fp32 WMMA GEMM: out = agg @ W^T + b ----------------
// Each wave computes a 16x32 strip of out (two 16x16 tiles, two independent
// accumulators so back-to-back v_wmma_f32_16x16x4_f32 can co-execute).
// A-frag (16x4 f32): lanes 0-15 hold M=lane, K={k,k+1}; lanes 16-31 K={k+2,k+3}.
// B-frag (4x16 f32): same striping with N=lane&15; B[k][n] = W[n][k] (W^T).
__global__ void __launch_bounds__(256)
k_gemm(const float* __restrict__ agg, const float* __restrict__ Wm,
       const float* __restrict__ bias, float* __restrict__ out, int M) {
  const int lane = threadIdx.x & 31;
  const int wave = threadIdx.x >> 5;
  const int pairsPerRow = C_DIM / 32;  // 4 pairs of 16-wide N tiles
  long wtile = (long)blockIdx.x * 8 + wave;
  int tileM = (int)(wtile / pairsPerRow);
  int tileNP = (int)(wtile % pairsPerRow);
  if (tileM * 16 >= M) return;  // wave-uniform: EXEC stays all-ones for WMMA
  const int m0 = tileM * 16;
  const int n0 = tileNP * 32;

  const int mr = lane & 15;          // row within tile (A) / N col (B)
  const int kb = (lane >> 4) << 1;   // K sub-offset: 0 or 2

  const float* arow = agg + (size_t)(m0 + mr) * C_DIM + kb;
  const float* w0 = Wm + (size_t)(n0 + mr) * C_DIM + kb;        // B tile 0
  const float* w1 = Wm + (size_t)(n0 + 16 + mr) * C_DIM + kb;   // B tile 1

  v8f acc0 = {};
  v8f acc1 = {};
#pragma unroll 8
  for (int k = 0; k < C_DIM; k += 4) {
    v2f a;  a.x  = arow[k]; a.y  = arow[k + 1];
    v2f b0; b0.x = w0[k];   b0.y = w0[k + 1];
    v2f b1; b1.x = w1[k];   b1.y = w1[k + 1];
    acc0 = __builtin_amdgcn_wmma_f32_16x16x4_f32(false, a, false, b0,
                                                 (short)0, acc0, false, false);
    acc1 = __builtin_amdgcn_wmma_f32_16x16x4_f32(false, a, false, b1,
                                                 (short)0, acc1, false, false);
  }

  // C/D layout: VGPR r -> M = r + 8*(lane>=16), N = lane&15
  const int n = lane & 15;
  const int mh = (lane >> 4) << 3;
  float bb0 = bias[n0 + n];
  float bb1 = bias[n0 + 16 + n];
  float* o = out + (size_t)(m0 + mh) * C_DIM + n0 + n;
#pragma unroll
  for (int r = 0; r < 8; ++r) {
    o[(size_t)r * C_DIM]      = acc0[r] + bb0;
    o[(size_t)r * C_DIM + 16] = acc1[r] + bb1;
  }
}

// ---------------- launch ----------------

extern "C" void kernel_launch(void* const* d_in, const int* in_sizes, int n_in,
                              void* d_out, int out_size, void* d_ws, size_t ws_size,
                              hipStream_t stream) {
  const float* x   = (const float*)d_in[0];       // [N, 128]
  const int*   ei  = (const int*)d_in[1];         // [2, E]
  const float* W   = (const float*)d_in[2];       // [128, 128]
  const float* b   = (const float*)d_in[3];       // [128]
  float* out = (float*)d_out;                     // [N, 128]

  const int Nn = in_sizes[0] / C_DIM;             // 50000
  const int E  = in_sizes[1] / 2;                 // 800000
  const int* row = ei;
  const int* col = ei + E;

  // workspace layout: agg [Nn*128] | deg [Nn] | dinv [Nn]
  float* agg  = (float*)d_ws;
  float* deg  = agg + (size_t)Nn * C_DIM;
  float* dinv = deg + Nn;

  const int T = 256;

  k_init_deg<<<(Nn + T - 1) / T, T, 0, stream>>>(deg, Nn);
  k_count_deg<<<(E + T - 1) / T, T, 0, stream>>>(col, deg, E);
  k_rsqrt<<<(Nn + T - 1) / T, T, 0, stream>>>(deg, dinv, Nn);

  int total = Nn * C_DIM;
  k_selfloop<<<(total + T - 1) / T, T, 0, stream>>>(x, dinv, agg, total);

  long scatterThreads = (long)E * 32;
  k_scatter<<<(int)((scatterThreads + T - 1) / T), T, 0, stream>>>(
      x, row, col, dinv, agg, E);

  int tilesM = (Nn + 15) / 16;
  long wtiles = (long)tilesM * (C_DIM / 32);
  int gemmBlocks = (int)((wtiles + 7) / 8);
  k_gemm<<<gemmBlocks, T, 0, stream>>>(agg, W, b, out, Nn);
}